// GCNModel_27711128994063
// MI455X (gfx1250) — compile-verified
//
#include <hip/hip_runtime.h>

typedef __attribute__((ext_vector_type(2))) float v2f;
typedef __attribute__((ext_vector_type(8))) float v8f;

__device__ __forceinline__ void atomicAddF32(float* p, float v) {
    (void)__hip_atomic_fetch_add(p, v, __ATOMIC_RELAXED, __HIP_MEMORY_SCOPE_AGENT);
}

// ---------------- degree / normalization ----------------

__global__ void k_deg_init(float* __restrict__ deg, int n) {
    int i = blockIdx.x * blockDim.x + threadIdx.x;
    if (i < n) deg[i] = 1.0f;  // self-loop contributes 1
}

__global__ void k_deg_accum(const int* __restrict__ col, float* __restrict__ deg, int E) {
    int i = blockIdx.x * blockDim.x + threadIdx.x;
    if (i < E) atomicAddF32(&deg[col[i]], 1.0f);
}

__global__ void k_rsqrt_inplace(float* __restrict__ deg, int n) {
    int i = blockIdx.x * blockDim.x + threadIdx.x;
    if (i < n) {
        float d = deg[i];
        deg[i] = (d > 0.0f) ? rsqrtf(d) : 0.0f;
    }
}

__global__ void k_edge_norm(const int* __restrict__ ei, const float* __restrict__ dinv,
                            float* __restrict__ norm, int E) {
    int i = blockIdx.x * blockDim.x + threadIdx.x;
    if (i < E) norm[i] = dinv[ei[i]] * dinv[ei[E + i]];
}

// ---------------- dense GEMM via V_WMMA_F32_16X16X4_F32 ----------------
// H[M,N] = act(A)[M,K] @ W[K,N].  If bias != nullptr, act(a) = relu(a + bias[col])
// (fused epilogue of the previous GCN layer).
// One block = 32 rows of A (two 16-row M-tiles). Wave w owns columns
// [16w, 16w+16) and computes BOTH M-tiles, so each B fragment (2 global
// loads) feeds two WMMAs -> 2 mem instructions per WMMA instead of 3.

__global__ __launch_bounds__(256) void k_gemm_wmma(
    const float* __restrict__ A, const float* __restrict__ W,
    const float* __restrict__ bias, float* __restrict__ H,
    int M, int K, int N)
{
    extern __shared__ float sA[];          // 32 x (K+4), padded vs bank conflicts
    const int tid  = threadIdx.x;
    const int row0 = blockIdx.x * 32;
    const int ldsS = K + 4;
    const int kvec = K >> 2;

    // --- stage 32xK A tile into LDS (coalesced float4), fold relu(a+bias)
    for (int i4 = tid; i4 < 32 * kvec; i4 += blockDim.x) {
        int r  = i4 / kvec;
        int c4 = i4 - r * kvec;
        int rg = row0 + r; if (rg >= M) rg = M - 1;
        float4 v = ((const float4*)(A + (size_t)rg * K))[c4];
        if (bias) {
            int c = c4 << 2;
            v.x = fmaxf(v.x + bias[c + 0], 0.0f);
            v.y = fmaxf(v.y + bias[c + 1], 0.0f);
            v.z = fmaxf(v.z + bias[c + 2], 0.0f);
            v.w = fmaxf(v.w + bias[c + 3], 0.0f);
        }
        ((float4*)(sA + r * ldsS))[c4] = v;
    }
    __syncthreads();

    const int lane  = tid & 31;
    const int wid   = tid >> 5;
    const int n0    = wid << 4;
    if (n0 >= N) return;                    // uniform per wave
    const int m     = lane & 15;            // A row within tile
    const int khalf = (lane >> 4) << 1;     // lanes 16-31 hold K+2,K+3
    const int ncol  = n0 + (lane & 15);     // B/D column

    v8f acc0 = {};
    v8f acc1 = {};
    #pragma unroll 4
    for (int k0 = 0; k0 < K; k0 += 4) {
        const int kk = k0 + khalf;
        v2f b, a0, a1;
        b.x  = W[(size_t)kk * N + ncol];
        b.y  = W[(size_t)(kk + 1) * N + ncol];
        a0.x = sA[m * ldsS + kk];
        a0.y = sA[m * ldsS + kk + 1];
        a1.x = sA[(m + 16) * ldsS + kk];
        a1.y = sA[(m + 16) * ldsS + kk + 1];
        acc0 = __builtin_amdgcn_wmma_f32_16x16x4_f32(
                   false, a0, false, b, (short)0, acc0, false, false);
        acc1 = __builtin_amdgcn_wmma_f32_16x16x4_f32(
                   false, a1, false, b, (short)0, acc1, false, false);
    }

    // D layout: VGPR v -> row v (lanes 0-15) / row v+8 (lanes 16-31)
    const int mbase = row0 + ((lane >> 4) << 3);
    #pragma unroll
    for (int v = 0; v < 8; ++v) {
        int rg0 = mbase + v;
        if (rg0 < M)      H[(size_t)rg0 * N + ncol]        = acc0[v];
        int rg1 = mbase + 16 + v;
        if (rg1 < M)      H[(size_t)rg1 * N + ncol]        = acc1[v];
    }
}

// ---------------- aggregation ----------------
// init: agg[i][c] = dinv[i]^2 * h[i][c] (+ bias[c])   (self-loop term, fused bias)

__global__ void k_agg_init(const float* __restrict__ h, const float* __restrict__ dinv,
                           const float* __restrict__ bias, float* __restrict__ agg,
                           int n, int C) {
    long long idx = (long long)blockIdx.x * blockDim.x + threadIdx.x;
    long long total = (long long)n * C;
    if (idx >= total) return;
    int i = (int)(idx / C);
    int c = (int)(idx - (long long)i * C);
    float d = dinv[i];
    float v = d * d * h[idx];
    if (bias) v += bias[c];
    agg[idx] = v;
}

// scatter: agg[col_e][c] += norm[e] * h[row_e][c]  (L2-resident f32 atomics)

__global__ void k_edge_scatter(const int* __restrict__ ei, const float* __restrict__ norm,
                               const float* __restrict__ h, float* __restrict__ agg,
                               int E, int C) {
    long long idx = (long long)blockIdx.x * blockDim.x + threadIdx.x;
    long long total = (long long)E * C;
    if (idx >= total) return;
    int e = (int)(idx / C);
    int c = (int)(idx - (long long)e * C);
    int r = ei[e];
    int t = ei[E + e];
    float v = norm[e] * h[(size_t)r * C + c];
    atomicAddF32(&agg[(size_t)t * C + c], v);
}

// ---------------- driver ----------------

extern "C" void kernel_launch(void* const* d_in, const int* in_sizes, int n_in,
                              void* d_out, int out_size, void* d_ws, size_t ws_size,
                              hipStream_t stream) {
    const float* x  = (const float*)d_in[0];
    const int*   ei = (const int*)  d_in[1];   // [2, E] int32 (row; col)
    const float* W1 = (const float*)d_in[2];
    const float* b1 = (const float*)d_in[3];
    const float* W2 = (const float*)d_in[4];
    const float* b2 = (const float*)d_in[5];
    float* out = (float*)d_out;

    const int E   = in_sizes[1] / 2;
    const int HID = in_sizes[3];               // 128
    const int OUT = in_sizes[5];               // 64
    const int IN  = in_sizes[2] / HID;         // 256
    const int N   = in_sizes[0] / IN;          // 100000

    // workspace carve-out (256B aligned)
    char* ws = (char*)d_ws;
    auto carve = [&](size_t bytes) -> float* {
        float* p = (float*)ws;
        ws += (bytes + 255) & ~(size_t)255;
        return p;
    };
    float* dinv = carve((size_t)N * sizeof(float));          // deg -> dinv in place
    float* enrm = carve((size_t)E * sizeof(float));
    float* h1   = carve((size_t)N * HID * sizeof(float));
    float* agg1 = carve((size_t)N * HID * sizeof(float));
    float* h2   = carve((size_t)N * OUT * sizeof(float));
    (void)ws_size; (void)n_in; (void)out_size;

    const int T = 256;
    auto blocks = [&](long long total) { return (unsigned)((total + T - 1) / T); };

    // symmetric normalization coefficients
    k_deg_init      <<<blocks(N), T, 0, stream>>>(dinv, N);
    k_deg_accum     <<<blocks(E), T, 0, stream>>>(ei + E, dinv, E);
    k_rsqrt_inplace <<<blocks(N), T, 0, stream>>>(dinv, N);
    k_edge_norm     <<<blocks(E), T, 0, stream>>>(ei, dinv, enrm, E);

    const unsigned mblocks = (unsigned)((N + 31) / 32);

    // layer 1: h1 = x @ W1
    k_gemm_wmma<<<mblocks, dim3((HID / 16) * 32), 32 * (IN + 4) * sizeof(float), stream>>>(
        x, W1, nullptr, h1, N, IN, HID);
    // agg1 = D^-1/2 A_hat D^-1/2 h1
    k_agg_init    <<<blocks((long long)N * HID), T, 0, stream>>>(h1, dinv, nullptr, agg1, N, HID);
    k_edge_scatter<<<blocks((long long)E * HID), T, 0, stream>>>(ei, enrm, h1, agg1, E, HID);

    // layer 2: h2 = relu(agg1 + b1) @ W2   (epilogue fused into A-tile load)
    k_gemm_wmma<<<mblocks, dim3((OUT / 16) * 32), 32 * (HID + 4) * sizeof(float), stream>>>(
        agg1, W2, b1, h2, N, HID, OUT);
    // out = b2 + D^-1/2 A_hat D^-1/2 h2   (bias fused into self-loop init)
    k_agg_init    <<<blocks((long long)N * OUT), T, 0, stream>>>(h2, dinv, b2, out, N, OUT);
    k_edge_scatter<<<blocks((long long)E * OUT), T, 0, stream>>>(ei, enrm, h2, out, E, OUT);
}